// SamplingMattersLoss_12395275616919
// MI455X (gfx1250) — compile-verified
//
#include <hip/hip_runtime.h>

// ---------------------------------------------------------------------------
// SamplingMatters margin triplet loss, CDNA5 (gfx1250, wave32).
//
// d_ap^2 for 16 triplets at once = diag( Q x Q^T ), Q[m][k] = a_m[k]-p_m[k],
// accumulated with V_WMMA_F32_16X16X4_F32. For this shape the A(16x4) and
// B(4x16 of Q^T) per-lane layouts coincide: lane L supplies
// Q[L%16][kbase + 2*(L/16) + {0,1}] for BOTH operands, so one float2 of Q
// feeds SRC0 and SRC1 of the same WMMA. Memory-minimal: each of the 16x3
// gathered rows is read exactly once per wave (8B/lane/chunk, L2-resident).
// ---------------------------------------------------------------------------

typedef __attribute__((ext_vector_type(2))) float v2f;
typedef __attribute__((ext_vector_type(8))) float v8f;

#define MARGIN_F   0.2f
#define EPS_F      1e-8f
#define D_DIM      512
#define WAVES_PB   8      // 256-thread block = 8 wave32
#define TRIPS_PW   16     // one 16x16 WMMA tile of triplets per wave

__global__ void sm_init_ws(float* ws) {
    if (threadIdx.x < 2) ws[threadIdx.x] = 0.0f;
}

__global__ __launch_bounds__(256) void sm_triplet_wmma(
        const float* __restrict__ emb,      // [B, 512] f32
        const int*   __restrict__ classes,  // [B] i32
        const int*   __restrict__ trips,    // [T, 3] i32
        const float* __restrict__ beta,     // [C] f32
        float*       __restrict__ ws,       // ws[0]=total, ws[1]=count
        int T)
{
    const int lane  = threadIdx.x & 31;
    const int wave  = threadIdx.x >> 5;
    const int gwave = blockIdx.x * WAVES_PB + wave;
    const int base  = gwave * TRIPS_PW;

    const int m    = lane & 15;   // triplet row within tile / matrix row M
    const int half = lane >> 4;   // K-half ownership for the 16x4 operand

    int t = base + m;
    if (t >= T) t = T - 1;        // clamp; masked out of the loss later

    const int ia = trips[t * 3 + 0];
    const int ip = trips[t * 3 + 1];
    const int in_ = trips[t * 3 + 2];

    const float* __restrict__ pa = emb + (size_t)ia * D_DIM;
    const float* __restrict__ pp = emb + (size_t)ip * D_DIM;
    const float* __restrict__ pn = emb + (size_t)in_ * D_DIM;

    v8f cap = {};   // Gram accumulator for (a-p)
    v8f can = {};   // Gram accumulator for (a-n)

    const int kb = 2 * half;      // this lane's K offset inside each 4-chunk
#pragma unroll 4
    for (int kk = 0; kk < D_DIM; kk += 4) {
        const int k0 = kk + kb;
        v2f av = *(const v2f*)(pa + k0);
        v2f pv = *(const v2f*)(pp + k0);
        v2f nv = *(const v2f*)(pn + k0);
        v2f qap = av - pv;
        v2f qan = av - nv;
        // D = Q * Q^T + C : same registers as SRC0 and SRC1 (symmetric layout)
        cap = __builtin_amdgcn_wmma_f32_16x16x4_f32(
                  false, qap, false, qap, (short)0, cap, false, false);
        can = __builtin_amdgcn_wmma_f32_16x16x4_f32(
                  false, qan, false, qan, (short)0, can, false, false);
    }

    // ---- Extract diagonal: element (m,m) lives in VGPR v, lane s, where
    //      m<8 : v=m,   s=m        (lanes 0-15 hold M=v)
    //      m>=8: v=m-8, s=m+16     (lanes 16-31 hold M=v+8)
    const int srcLane = (m < 8) ? m : (m + 16);
    float dap2 = 0.0f, dan2 = 0.0f;
#pragma unroll
    for (int v = 0; v < 8; ++v) {
        float t0 = __shfl(cap[v], srcLane, 32);
        float t1 = __shfl(can[v], srcLane, 32);
        if ((m & 7) == v) { dap2 = t0; dan2 = t1; }
    }

    // ---- Loss on lanes 0-15 (one triplet each); lanes 16-31 contribute 0.
    float contrib = 0.0f, cnt = 0.0f;
    if (half == 0 && (base + m) < T) {
        float d_ap = sqrtf(dap2 + EPS_F);
        float d_an = sqrtf(dan2 + EPS_F);
        float b    = beta[classes[ia]];
        float pos  = fmaxf(d_ap - b + MARGIN_F, 0.0f);
        float neg  = fmaxf(b - d_an + MARGIN_F, 0.0f);
        contrib = pos + neg;
        cnt     = (pos > 0.0f || neg > 0.0f) ? 1.0f : 0.0f;
    }

    // ---- Wave32 reduction, then one pair of f32 atomics per wave.
#pragma unroll
    for (int off = 16; off >= 1; off >>= 1) {
        contrib += __shfl_xor(contrib, off, 32);
        cnt     += __shfl_xor(cnt, off, 32);
    }
    if (lane == 0) {
        atomicAdd(&ws[0], contrib);
        atomicAdd(&ws[1], cnt);
    }
}

__global__ void sm_finalize(const float* __restrict__ ws,
                            float* __restrict__ out)
{
    float total = ws[0];
    float cnt   = ws[1];
    out[0] = (cnt == 0.0f) ? total : total / fmaxf(cnt, 1.0f);
}

extern "C" void kernel_launch(void* const* d_in, const int* in_sizes, int n_in,
                              void* d_out, int out_size, void* d_ws, size_t ws_size,
                              hipStream_t stream)
{
    const float* emb     = (const float*)d_in[0];   // [8192*512]
    const int*   classes = (const int*)  d_in[1];   // [8192]
    const int*   trips   = (const int*)  d_in[2];   // [T*3]
    const float* beta    = (const float*)d_in[3];   // [1000]
    float*       out     = (float*)d_out;
    float*       ws      = (float*)d_ws;            // 2 floats: total, count

    const int T = in_sizes[2] / 3;
    const int trips_per_block = TRIPS_PW * WAVES_PB;        // 128
    const int grid = (T + trips_per_block - 1) / trips_per_block;

    sm_init_ws<<<1, 64, 0, stream>>>(ws);
    sm_triplet_wmma<<<grid, 256, 0, stream>>>(emb, classes, trips, beta, ws, T);
    sm_finalize<<<1, 1, 0, stream>>>(ws, out);
}